// Block_82403242541237
// MI455X (gfx1250) — compile-verified
//
#include <hip/hip_runtime.h>
#include <math.h>

// ---------------------------------------------------------------------------
// PointNet-style residual block, fully fused for MI455X (gfx1250, wave32).
//   - All GEMMs via V_WMMA_F32_16X16X4_F32 (exact fp32 math).
//   - Weight B-fragments (except W2) are register-resident (loaded once per wave from
//     global/L2): those GEMM loops issue only A-fragment ds_loads (W2 is read from LDS).
//   - Gather+MLP fused: the gathered [n*k,64] matrix is never materialized;
//     x (25.6MB) stays L2-resident, neighbors are re-gathered per BN pass.
//   - BatchNorm batch stats via per-lane register accumulation -> LDS
//     ds_add_f32 -> one global atomic per channel per block.
//   - MLP biases dropped (they cancel inside BatchNorm).
// Workspace layout (floats): [0,2048) stats/scales, then xbuf[n*64], mbuf[n*64].
// ---------------------------------------------------------------------------

typedef float v2f __attribute__((ext_vector_type(2)));
typedef float v8f __attribute__((ext_vector_type(8)));

#define BN_EPS 1e-5f

// Load a 16x4 fp32 WMMA fragment from a row-major [16+][64] LDS tile.
// Lane L: row = L%16, K-pair = 4*j + 2*(L/16)  (per CDNA5 ISA 7.12.2 layout).
__device__ __forceinline__ v2f frag_ld(const float* base, int lane, int j) {
    int row = lane & 15;
    int kk  = (j << 2) + ((lane >> 4) << 1);
    const float* p = base + row * 64 + kk;
    v2f r; r.x = p[0]; r.y = p[1];
    return r;
}

// Load a 4x16 B-fragment of W^T directly from global weight W[64(out)][64(in)]:
// B[k][o] = W[o][k]; lane L: o = 16*t + L%16, k-pair = 4*j + 2*(L/16).
__device__ __forceinline__ v2f frag_ld_w(const float* __restrict__ W,
                                         int lane, int j, int t) {
    int o  = (t << 4) + (lane & 15);
    int kk = (j << 2) + ((lane >> 4) << 1);
    const float* p = W + o * 64 + kk;
    v2f r; r.x = p[0]; r.y = p[1];
    return r;
}

__device__ __forceinline__ v8f wmma4(v2f a, v2f b, v8f c) {
    // D = A(16x4) * B(4x16) + C(16x16), fp32
    return __builtin_amdgcn_wmma_f32_16x16x4_f32(false, a, false, b,
                                                 (short)0, c, false, false);
}

// ---------------------------------------------------------------------------
// Dense GEMM: Z[nrows,64] = X[nrows,64] @ W^T, W row-major [64(out)][64(in)].
// Also accumulates per-channel sum / sumsq of Z (pre-BN stats).
// One wave per 16-row tile; 4 waves / block; W fragments register-resident.
// Safe when Z == X (row-wise in-place).
// ---------------------------------------------------------------------------
__global__ void __launch_bounds__(128)
gemm64_stats(const float* __restrict__ X, const float* __restrict__ W,
             float* __restrict__ Z, float* __restrict__ sumOut,
             float* __restrict__ sqOut, int nrows)
{
    __shared__ float tiles[4][16 * 64];
    __shared__ float sSum[64], sSq[64];

    int tid = threadIdx.x;
    if (tid < 64) { sSum[tid] = 0.f; sSq[tid] = 0.f; }
    __syncthreads();

    int lane = tid & 31, wave = tid >> 5;
    float* tile = tiles[wave];

    // register-resident weight fragments (64 x v2f = 128 VGPRs)
    v2f bf[16][4];
    #pragma unroll
    for (int j = 0; j < 16; ++j)
        #pragma unroll
        for (int t = 0; t < 4; ++t)
            bf[j][t] = frag_ld_w(W, lane, j, t);

    int ntiles = (nrows + 15) >> 4;
    int wstep  = gridDim.x << 2;

    float sum[4] = {0, 0, 0, 0}, sq[4] = {0, 0, 0, 0};

    for (int tIdx = (blockIdx.x << 2) + wave; tIdx < ntiles; tIdx += wstep) {
        int m0 = tIdx << 4;
        // stage 16 rows, coalesced (2 rows per b128 wave-load)
        #pragma unroll
        for (int i = 0; i < 8; ++i) {
            int r   = (i << 1) + (lane >> 4);
            int row = m0 + r;
            float4 v = {0.f, 0.f, 0.f, 0.f};
            if (row < nrows)
                v = *(const float4*)(X + (size_t)row * 64 + ((lane & 15) << 2));
            *(float4*)(tile + r * 64 + ((lane & 15) << 2)) = v;
        }
        __builtin_amdgcn_wave_barrier();

        v8f acc[4] = {};
        #pragma unroll
        for (int j = 0; j < 16; ++j) {
            v2f a = frag_ld(tile, lane, j);   // only LDS traffic in the loop
            #pragma unroll
            for (int t = 0; t < 4; ++t)
                acc[t] = wmma4(a, bf[j][t], acc[t]);
        }

        #pragma unroll
        for (int t = 0; t < 4; ++t) {
            int ch = (t << 4) + (lane & 15);
            #pragma unroll
            for (int e = 0; e < 8; ++e) {
                int row = m0 + e + ((lane >> 4) << 3);
                float v = acc[t][e];
                if (row < nrows) {
                    Z[(size_t)row * 64 + ch] = v;
                    sum[t] += v; sq[t] += v * v;
                }
            }
        }
        __builtin_amdgcn_wave_barrier();
    }

    #pragma unroll
    for (int t = 0; t < 4; ++t) {
        int ch = (t << 4) + (lane & 15);
        atomicAdd(&sSum[ch], sum[t]);
        atomicAdd(&sSq[ch], sq[t]);
    }
    __syncthreads();
    if (tid < 64) { atomicAdd(&sumOut[tid], sSum[tid]); atomicAdd(&sqOut[tid], sSq[tid]); }
}

// ---------------------------------------------------------------------------
// Fused LocalAggregation. One wave == one point (its k=16 neighbors are the
// 16 GEMM rows). W1 fragments register-resident; W2 streamed from LDS.
// MODE 0: gather+GEMM1, stats(h1). MODE 1: +BN1+ReLU+GEMM2, stats(h2).
// MODE 2: +BN2+ReLU+maxpool -> Mout, stats(m).
// ---------------------------------------------------------------------------
template <int MODE>
__global__ void __launch_bounds__(128)
fused_local_agg(const float* __restrict__ X, const int* __restrict__ RIDX,
                const float* __restrict__ W1, const float* __restrict__ W2,
                const float* __restrict__ sc1, const float* __restrict__ sh1,
                const float* __restrict__ sc2, const float* __restrict__ sh2,
                float* __restrict__ Mout, float* __restrict__ sumOut,
                float* __restrict__ sqOut, int npoints)
{
    __shared__ float w2lds[64 * 64];
    __shared__ float tiles[4][16 * 64];
    __shared__ float sSum[64], sSq[64];

    int tid = threadIdx.x;
    if (MODE >= 1)
        for (int i = tid; i < 64 * 64 / 4; i += 128)
            ((float4*)w2lds)[i] = ((const float4*)W2)[i];
    if (tid < 64) { sSum[tid] = 0.f; sSq[tid] = 0.f; }
    __syncthreads();

    int lane = tid & 31, wave = tid >> 5;
    float* tile = tiles[wave];

    // register-resident W1 fragments (128 VGPRs)
    v2f bf1[16][4];
    #pragma unroll
    for (int j = 0; j < 16; ++j)
        #pragma unroll
        for (int t = 0; t < 4; ++t)
            bf1[j][t] = frag_ld_w(W1, lane, j, t);

    // per-lane BN constants (channel = 16*t + lane%16, fixed per lane)
    float s1[4], h1[4], s2[4], h2[4];
    #pragma unroll
    for (int t = 0; t < 4; ++t) {
        int ch = (t << 4) + (lane & 15);
        if (MODE >= 1) { s1[t] = sc1[ch]; h1[t] = sh1[ch]; }
        if (MODE >= 2) { s2[t] = sc2[ch]; h2[t] = sh2[ch]; }
    }

    float sum[4] = {0, 0, 0, 0}, sq[4] = {0, 0, 0, 0};
    int wstep = gridDim.x << 2;

    for (int p = (blockIdx.x << 2) + wave; p < npoints; p += wstep) {
        // gather 16 neighbor rows of x into the wave's LDS tile (L2-resident)
        #pragma unroll
        for (int i = 0; i < 8; ++i) {
            int r    = (i << 1) + (lane >> 4);
            int rowi = RIDX[(size_t)p * 16 + r];
            float4 v = *(const float4*)(X + (size_t)rowi * 64 + ((lane & 15) << 2));
            *(float4*)(tile + r * 64 + ((lane & 15) << 2)) = v;
        }
        __builtin_amdgcn_wave_barrier();

        // GEMM1: h1 = grouped @ W1^T   (64 x v_wmma_f32_16x16x4_f32,
        // B operands from registers; only A fragments hit LDS)
        v8f acc[4] = {};
        #pragma unroll
        for (int j = 0; j < 16; ++j) {
            v2f a = frag_ld(tile, lane, j);
            #pragma unroll
            for (int t = 0; t < 4; ++t)
                acc[t] = wmma4(a, bf1[j][t], acc[t]);
        }

        if (MODE == 0) {
            #pragma unroll
            for (int t = 0; t < 4; ++t)
                #pragma unroll
                for (int e = 0; e < 8; ++e) { float v = acc[t][e]; sum[t] += v; sq[t] += v * v; }
            continue;
        }

        // BN1 + ReLU, transpose C-layout -> A-layout through the (reused) tile
        __builtin_amdgcn_wave_barrier();
        #pragma unroll
        for (int t = 0; t < 4; ++t) {
            int ch = (t << 4) + (lane & 15);
            #pragma unroll
            for (int e = 0; e < 8; ++e) {
                int M = e + ((lane >> 4) << 3);
                tile[M * 64 + ch] = fmaxf(acc[t][e] * s1[t] + h1[t], 0.f);
            }
        }
        __builtin_amdgcn_wave_barrier();

        // GEMM2: h2 = h1 @ W2^T  (B from LDS)
        v8f acc2[4] = {};
        #pragma unroll
        for (int j = 0; j < 16; ++j) {
            v2f a = frag_ld(tile, lane, j);
            #pragma unroll
            for (int t = 0; t < 4; ++t)
                acc2[t] = wmma4(a, frag_ld(w2lds + (t << 4) * 64, lane, j), acc2[t]);
        }

        if (MODE == 1) {
            #pragma unroll
            for (int t = 0; t < 4; ++t)
                #pragma unroll
                for (int e = 0; e < 8; ++e) { float v = acc2[t][e]; sum[t] += v; sq[t] += v * v; }
            continue;
        }

        // BN2 + ReLU + max-pool over the 16 neighbor rows
        #pragma unroll
        for (int t = 0; t < 4; ++t) {
            float mv = 0.f;  // ReLU output is >= 0
            #pragma unroll
            for (int e = 0; e < 8; ++e)
                mv = fmaxf(mv, fmaxf(acc2[t][e] * s2[t] + h2[t], 0.f));
            mv = fmaxf(mv, __shfl_xor(mv, 16, 32));  // other 8 rows
            if (lane < 16) {
                Mout[(size_t)p * 64 + (t << 4) + lane] = mv;
                sum[t] += mv; sq[t] += mv * mv;
            }
        }
        __builtin_amdgcn_wave_barrier();
    }

    #pragma unroll
    for (int t = 0; t < 4; ++t) {
        if (MODE == 2 && lane >= 16) break;  // only lanes<16 hold pooled stats
        int ch = (t << 4) + (lane & 15);
        atomicAdd(&sSum[ch], sum[t]);
        atomicAdd(&sSq[ch], sq[t]);
    }
    __syncthreads();
    if (tid < 64) { atomicAdd(&sumOut[tid], sSum[tid]); atomicAdd(&sqOut[tid], sSq[tid]); }
}

// sum/sumsq -> BN scale/shift (scale = g*rsqrt(var+eps), shift = b - mean*scale)
__global__ void bn_finalize(const float* __restrict__ sum, const float* __restrict__ sq,
                            float invCount, const float* __restrict__ g,
                            const float* __restrict__ b, float* __restrict__ scale,
                            float* __restrict__ shift)
{
    int c = threadIdx.x;
    if (c < 64) {
        float mean = sum[c] * invCount;
        float var  = sq[c] * invCount - mean * mean;
        float s    = g[c] * rsqrtf(var + BN_EPS);
        scale[c] = s;
        shift[c] = b[c] - mean * s;
    }
}

// in-place y = relu(y*scale[c] + shift[c]), c = column index (row length 64)
__global__ void apply_bn_relu(float* __restrict__ buf, const float* __restrict__ scale,
                              const float* __restrict__ shift, int total4)
{
    int i = blockIdx.x * blockDim.x + threadIdx.x;
    if (i >= total4) return;
    float4 v = ((float4*)buf)[i];
    int c = (i << 2) & 63;
    v.x = fmaxf(v.x * scale[c + 0] + shift[c + 0], 0.f);
    v.y = fmaxf(v.y * scale[c + 1] + shift[c + 1], 0.f);
    v.z = fmaxf(v.z * scale[c + 2] + shift[c + 2], 0.f);
    v.w = fmaxf(v.w * scale[c + 3] + shift[c + 3], 0.f);
    ((float4*)buf)[i] = v;
}

// out = relu(feat + z3*scale[c] + shift[c])   (residual + BN3 + ReLU)
__global__ void final_residual(const float* __restrict__ feat, const float* __restrict__ z3,
                               const float* __restrict__ scale, const float* __restrict__ shift,
                               float* __restrict__ out, int total4)
{
    int i = blockIdx.x * blockDim.x + threadIdx.x;
    if (i >= total4) return;
    float4 f = ((const float4*)feat)[i];
    float4 z = ((const float4*)z3)[i];
    int c = (i << 2) & 63;
    float4 o;
    o.x = fmaxf(f.x + z.x * scale[c + 0] + shift[c + 0], 0.f);
    o.y = fmaxf(f.y + z.y * scale[c + 1] + shift[c + 1], 0.f);
    o.z = fmaxf(f.z + z.z * scale[c + 2] + shift[c + 2], 0.f);
    o.w = fmaxf(f.w + z.w * scale[c + 3] + shift[c + 3], 0.f);
    ((float4*)out)[i] = o;
}

// ---------------------------------------------------------------------------
extern "C" void kernel_launch(void* const* d_in, const int* in_sizes, int n_in,
                              void* d_out, int out_size, void* d_ws, size_t ws_size,
                              hipStream_t stream)
{
    // inputs (setup_inputs order); coord(0) and mlp biases(5,7) unused
    const float* feat  = (const float*)d_in[1];
    const int*   ridx  = (const int*)d_in[2];     // reference_index [n,16]
    const float* fc1_w = (const float*)d_in[3];
    const float* w1    = (const float*)d_in[4];
    const float* w2    = (const float*)d_in[6];
    const float* fc3_w = (const float*)d_in[8];
    const float* g1  = (const float*)d_in[9],  *b1  = (const float*)d_in[10];
    const float* gm1 = (const float*)d_in[11], *bm1 = (const float*)d_in[12];
    const float* gm2 = (const float*)d_in[13], *bm2 = (const float*)d_in[14];
    const float* g2  = (const float*)d_in[15], *b2  = (const float*)d_in[16];
    const float* g3  = (const float*)d_in[17], *b3  = (const float*)d_in[18];

    int n = in_sizes[1] / 64;                     // 100000 (c = 64, k = 16)

    float* ws   = (float*)d_ws;
    float* SUM1 = ws + 0,    *SQ1 = ws + 64;
    float* SUMa = ws + 128,  *SQa = ws + 192;     // MLP layer-1 stats
    float* SUMb = ws + 256,  *SQb = ws + 320;     // MLP layer-2 stats
    float* SUM2 = ws + 384,  *SQ2 = ws + 448;     // pooled stats
    float* SUM3 = ws + 512,  *SQ3 = ws + 576;     // fc3 stats
    float* SC1  = ws + 640,  *SH1 = ws + 704;
    float* SCa  = ws + 768,  *SHa = ws + 832;
    float* SCb  = ws + 896,  *SHb = ws + 960;
    float* SC2  = ws + 1024, *SH2 = ws + 1088;
    float* SC3  = ws + 1152, *SH3 = ws + 1216;
    float* xbuf = ws + 2048;                      // [n,64] : z1 -> x
    float* mbuf = xbuf + (size_t)n * 64;          // [n,64] : m -> t -> z3

    float invN  = (float)(1.0 / (double)n);
    float invNK = (float)(1.0 / ((double)n * 16.0));
    int   e4    = n * 16;                         // n*64/4 float4 elements
    int   ebl   = (e4 + 255) / 256;

    hipMemsetAsync(ws, 0, 2048 * sizeof(float), stream);

    // fc1 -> BN1 -> ReLU  (x, L2-resident afterwards)
    gemm64_stats<<<512, 128, 0, stream>>>(feat, fc1_w, xbuf, SUM1, SQ1, n);
    bn_finalize<<<1, 64, 0, stream>>>(SUM1, SQ1, invN, g1, b1, SC1, SH1);
    apply_bn_relu<<<ebl, 256, 0, stream>>>(xbuf, SC1, SH1, e4);

    // fused gather+MLP, three BN-stat passes (recompute instead of spilling 400MB)
    fused_local_agg<0><<<1024, 128, 0, stream>>>(xbuf, ridx, w1, w2,
        nullptr, nullptr, nullptr, nullptr, nullptr, SUMa, SQa, n);
    bn_finalize<<<1, 64, 0, stream>>>(SUMa, SQa, invNK, gm1, bm1, SCa, SHa);

    fused_local_agg<1><<<1024, 128, 0, stream>>>(xbuf, ridx, w1, w2,
        SCa, SHa, nullptr, nullptr, nullptr, SUMb, SQb, n);
    bn_finalize<<<1, 64, 0, stream>>>(SUMb, SQb, invNK, gm2, bm2, SCb, SHb);

    fused_local_agg<2><<<1024, 128, 0, stream>>>(xbuf, ridx, w1, w2,
        SCa, SHa, SCb, SHb, mbuf, SUM2, SQ2, n);
    bn_finalize<<<1, 64, 0, stream>>>(SUM2, SQ2, invN, g2, b2, SC2, SH2);

    // BN2 -> ReLU -> fc3 (in-place, row-wise safe) -> BN3 -> residual -> ReLU
    apply_bn_relu<<<ebl, 256, 0, stream>>>(mbuf, SC2, SH2, e4);
    gemm64_stats<<<512, 128, 0, stream>>>(mbuf, fc3_w, mbuf, SUM3, SQ3, n);
    bn_finalize<<<1, 64, 0, stream>>>(SUM3, SQ3, invN, g3, b3, SC3, SH3);
    final_residual<<<ebl, 256, 0, stream>>>(feat, mbuf, SC3, SH3, (float*)d_out, e4);
}